// TopKRouter_2877628088575
// MI455X (gfx1250) — compile-verified
//
#include <hip/hip_runtime.h>
#include <hip/hip_bf16.h>
#include <math.h>
#include <stdint.h>

// TopK router: logits = x(T,1024) @ W(64,1024)^T in fp32 via V_WMMA_F32_16X16X4_F32,
// then per-token top-8 + softmax.
// Memory-bound: 256MB of x streamed @ 23.3 TB/s ~= 11us floor; fp32 WMMA keeps the
// 8.6 GFLOP of GEMM under that floor. This round: double-buffered LDS staging driven
// by the Tensor Data Mover (tensor_load_to_lds + s_wait_tensorcnt) so the global
// stream overlaps WMMA; TDM pad_enable reproduces the 68-float padded LDS rows.

typedef float v2f __attribute__((ext_vector_type(2)));
typedef float v4f __attribute__((ext_vector_type(4)));
typedef float v8f __attribute__((ext_vector_type(8)));
typedef unsigned int v4u __attribute__((ext_vector_type(4)));
typedef int v4i_t __attribute__((ext_vector_type(4)));
typedef int v8i_t __attribute__((ext_vector_type(8)));

#define DD   1024   // hidden dim
#define EE   64     // experts
#define TK   8      // top_k
#define KC   64     // K chunk staged in LDS
#define MT   32     // tokens per workgroup
#define XSTR 68     // LDS row stride (floats), 68 % 64 == 4 -> conflict-free b64 frag reads
#define WSTR 68
#define LSTR (EE + 1)
#define NCH  (DD / KC)

#if defined(__HIP_DEVICE_COMPILE__) && __has_builtin(__builtin_amdgcn_tensor_load_to_lds) && __has_builtin(__builtin_amdgcn_s_wait_tensorcnt)
#define USE_TDM 1
#else
#define USE_TDM 0
#endif

#if USE_TDM
// Issue one TDM 2D tile load: tile_w = KC (=64) floats wide, tile_h rows,
// row stride DD floats, destination LDS rows padded 64->68 DWORDs via TDM padding.
// D# layout per CDNA5 ISA ch.8 (groups 0/1; groups 2/3 zero for 2D tiles).
__device__ __forceinline__ void tdm_load_tile(unsigned lds_off, const float* gptr, int tile_h) {
    const uint64_t ga = (uint64_t)(uintptr_t)gptr;
    v4u g0;
    g0.x = 1u;                                   // count=1, user descriptor
    g0.y = lds_off;                              // lds_addr (bytes)
    g0.z = (unsigned)ga;                         // global_addr[31:0]
    g0.w = (unsigned)((ga >> 32) & 0x01FFFFFFu)  // global_addr[56:32]
           | (2u << 30);                         // type = 2 ("image")
    // group1: data_size=4B (2<<16), pad_enable (1<<20),
    // pad_interval=5 -> every 64 DWORDs (22), pad_amount=3 -> 4 DWORDs (25)
    const unsigned dw0 = (2u << 16) | (1u << 20) | (5u << 22) | (3u << 25);
    v8i_t g1;
    g1[0] = (int)dw0;
    g1[1] = (int)((unsigned)KC << 16);                     // tensor_dim0[15:0] @ bits[63:48]
    g1[2] = (int)(((unsigned)tile_h & 0xFFFFu) << 16);     // dim0[31:16]=0 | tensor_dim1[15:0]
    g1[3] = (int)((unsigned)KC << 16);                     // dim1[31:16]=0 | tile_dim0=KC
    g1[4] = (int)((unsigned)tile_h & 0xFFFFu);             // tile_dim1 | tile_dim2=0
    g1[5] = (int)DD;                                       // tensor_dim0_stride[31:0] = 1024
    g1[6] = 0;                                             // stride hi | dim1_stride lo
    g1[7] = 0;
    v4i_t z4 = {0, 0, 0, 0};
#if __clang_major__ >= 23
    v8i_t z8 = {0, 0, 0, 0, 0, 0, 0, 0};
    __builtin_amdgcn_tensor_load_to_lds(g0, g1, z4, z4, z8, 0);
#else
    __builtin_amdgcn_tensor_load_to_lds(g0, g1, z4, z4, 0);
#endif
}
#endif

// Fallback staging helpers (also used on host parse pass).
__device__ __forceinline__ void load_chunk_regs(const float* __restrict__ x,
                                                const float* __restrict__ W,
                                                int tok0, int k0, int tid,
                                                v4f xr[2], v4f wr[4]) {
    #pragma unroll
    for (int i = 0; i < 2; ++i) {
        int idx = tid + i * 256;
        int row = idx >> 4, col = (idx & 15) << 2;
        xr[i] = *(const v4f*)(x + (size_t)(tok0 + row) * DD + k0 + col);
    }
    #pragma unroll
    for (int i = 0; i < 4; ++i) {
        int idx = tid + i * 256;
        int row = idx >> 4, col = (idx & 15) << 2;
        wr[i] = *(const v4f*)(W + (size_t)row * DD + k0 + col);
    }
}
__device__ __forceinline__ void store_chunk_lds(float* __restrict__ xbuf,
                                                float* __restrict__ wbuf,
                                                int tid, const v4f xr[2], const v4f wr[4]) {
    #pragma unroll
    for (int i = 0; i < 2; ++i) {
        int idx = tid + i * 256;
        int row = idx >> 4, col = (idx & 15) << 2;
        *(v4f*)(&xbuf[row * XSTR + col]) = xr[i];
    }
    #pragma unroll
    for (int i = 0; i < 4; ++i) {
        int idx = tid + i * 256;
        int row = idx >> 4, col = (idx & 15) << 2;
        *(v4f*)(&wbuf[row * WSTR + col]) = wr[i];
    }
}

__global__ __launch_bounds__(256)
void topk_router_kernel(const float* __restrict__ x, const float* __restrict__ W,
                        int* __restrict__ out_idx, float* __restrict__ out_w, int T) {
    __shared__ float xs[2][MT * XSTR];   // double-buffered x chunk
    __shared__ float ws[2][EE * WSTR];   // double-buffered W chunk
    __shared__ float logits[MT * LSTR];

    const int tid  = threadIdx.x;
    const int wave = tid >> 5;
    const int lane = tid & 31;
    const int mt   = wave >> 2;       // 0..1 : 16-token sub-tile
    const int nt   = wave & 3;        // 0..3 : 16-expert sub-tile
    const int tok0 = blockIdx.x * MT;

    v8f acc = {0.f, 0.f, 0.f, 0.f, 0.f, 0.f, 0.f, 0.f};

    // f32 16x16x4 A/B fragment addressing: lanes 0-15 -> K pair {0,1}, 16-31 -> {2,3}
    const int kpair = (lane >> 4) << 1;
    const int arow  = mt * 16 + (lane & 15);
    const int brow  = nt * 16 + (lane & 15);

#if USE_TDM
    if (wave == 0) {
        tdm_load_tile((unsigned)(uintptr_t)&xs[0][0], x + (size_t)tok0 * DD, MT);
        tdm_load_tile((unsigned)(uintptr_t)&ws[0][0], W, EE);
    }
    for (int i = 0; i < NCH; ++i) {
        const int cur = i & 1;
        const int k1  = (i + 1) * KC;
        if (wave == 0 && k1 < DD) {   // prefetch next chunk into the other buffer
            tdm_load_tile((unsigned)(uintptr_t)&xs[cur ^ 1][0],
                          x + (size_t)tok0 * DD + k1, MT);
            tdm_load_tile((unsigned)(uintptr_t)&ws[cur ^ 1][0], W + k1, EE);
        }
        // TDM completes in order: with 2 newer descriptors in flight, <=2 means
        // this chunk's pair has landed in LDS.
        if (k1 < DD) __builtin_amdgcn_s_wait_tensorcnt(2);
        else         __builtin_amdgcn_s_wait_tensorcnt(0);
        __syncthreads();

        const float* xfrag = &xs[cur][arow * XSTR + kpair];
        const float* wfrag = &ws[cur][brow * WSTR + kpair];
        #pragma unroll
        for (int kk = 0; kk < KC; kk += 4) {
            v2f a = *(const v2f*)(xfrag + kk);
            v2f b = *(const v2f*)(wfrag + kk);
            acc = __builtin_amdgcn_wmma_f32_16x16x4_f32(
                false, a, false, b, (short)0, acc, false, false);
        }
        __syncthreads();  // all readers done before TDM overwrites this buffer
    }
#else
    {   // double-buffered fallback: regular global loads + ds stores
        v4f xr[2], wr[4];
        load_chunk_regs(x, W, tok0, 0, tid, xr, wr);
        store_chunk_lds(&xs[0][0], &ws[0][0], tid, xr, wr);
        __syncthreads();
        for (int i = 0; i < NCH; ++i) {
            const int cur = i & 1;
            const int k1  = (i + 1) * KC;
            if (k1 < DD) load_chunk_regs(x, W, tok0, k1, tid, xr, wr);

            const float* xfrag = &xs[cur][arow * XSTR + kpair];
            const float* wfrag = &ws[cur][brow * WSTR + kpair];
            #pragma unroll
            for (int kk = 0; kk < KC; kk += 4) {
                v2f a = *(const v2f*)(xfrag + kk);
                v2f b = *(const v2f*)(wfrag + kk);
                acc = __builtin_amdgcn_wmma_f32_16x16x4_f32(
                    false, a, false, b, (short)0, acc, false, false);
            }
            if (k1 < DD) store_chunk_lds(&xs[cur ^ 1][0], &ws[cur ^ 1][0], tid, xr, wr);
            __syncthreads();
        }
    }
#endif

    // Scatter C tile into logits LDS. C/D layout: VGPR v -> M = v (lanes 0-15),
    // M = 8+v (lanes 16-31); N = lane & 15.
    {
        const int n     = nt * 16 + (lane & 15);
        const int mbase = mt * 16 + ((lane >> 4) << 3);
        #pragma unroll
        for (int v = 0; v < 8; ++v)
            logits[(mbase + v) * LSTR + n] = acc[v];
    }
    __syncthreads();

    // Top-8 + softmax: one token per thread (first 32 threads).
    if (tid < MT) {
        float* row = &logits[tid * LSTR];
        float vals[TK];
        int   idxs[TK];
        #pragma unroll
        for (int k = 0; k < TK; ++k) {
            float best = -INFINITY;
            int   bi   = 0;
            #pragma unroll
            for (int e = 0; e < EE; ++e) {
                float v = row[e];
                if (v > best) { best = v; bi = e; }   // strict > : lowest index on ties
            }
            vals[k] = best;
            idxs[k] = bi;
            row[bi] = -INFINITY;  // dynamic LDS index, stays off scratch
        }
        float m   = vals[0];      // sorted descending
        float ex[TK];
        float sum = 0.f;
        #pragma unroll
        for (int k = 0; k < TK; ++k) { ex[k] = __expf(vals[k] - m); sum += ex[k]; }
        float inv = 1.0f / sum;
        size_t t = (size_t)(tok0 + tid);
        #pragma unroll
        for (int k = 0; k < TK; ++k) {
            out_idx[t * TK + k] = idxs[k];
            out_w[t * TK + k]   = ex[k] * inv;
        }
    }
}

extern "C" void kernel_launch(void* const* d_in, const int* in_sizes, int n_in,
                              void* d_out, int out_size, void* d_ws, size_t ws_size,
                              hipStream_t stream) {
    const float* x = (const float*)d_in[0];
    const float* W = (const float*)d_in[1];
    // d_in[2] = top_k scalar (hardcoded 8)
    const int T = in_sizes[0] / DD;   // 65536

    int*   out_idx = (int*)d_out;                       // first T*8 int32
    float* out_w   = (float*)d_out + (size_t)T * TK;    // then  T*8 float32

    dim3 grid((T + MT - 1) / MT);
    dim3 block(256);
    topk_router_kernel<<<grid, block, 0, stream>>>(x, W, out_idx, out_w, T);
}